// ColorConsistencyLoss_24653112279585
// MI455X (gfx1250) — compile-verified
//
#include <hip/hip_runtime.h>

#define K_SEG 64
#define BATCH 8
#define PIX   (1024 * 1024)
#define GX    128                 // blocks in x per image
#define NQ    (PIX / 4)           // 262144 quads per image
#define ITERS (NQ / (GX * 256))   // = 8, exact
#define HSTRIDE (K_SEG * 3)       // 192 floats per wave-private histogram

typedef __attribute__((ext_vector_type(2))) float v2f;
typedef __attribute__((ext_vector_type(8))) float v8f;

__device__ __forceinline__ float smooth_l1(float d) {
    float ad = fabsf(d);
    return (ad < 1.0f) ? 0.5f * d * d : ad - 0.5f;
}

// ---------------- pass 0: zero the segment-sum table -------------------------
__global__ void cc_zero_ws(float* __restrict__ sums) {
    int i = blockIdx.x * blockDim.x + threadIdx.x;
    if (i < BATCH * HSTRIDE) sums[i] = 0.0f;
}

// ---------------- pass 1: per-image segment sums/counts ----------------------
// sums layout: [b][k][{count, sumA, sumB}]
// Per-wave private LDS histograms: no inter-wave same-address contention,
// only intra-wave bank conflicts (k*3 mod 64 is a bijection in k).
__global__ __launch_bounds__(256) void cc_seg_sums(const float* __restrict__ pred,
                                                   const int* __restrict__ masks,
                                                   float* __restrict__ sums) {
    __shared__ float hist[8 * HSTRIDE];  // 8 waves x 192 floats = 6 KB
    const int tid  = threadIdx.x;
    const int b    = blockIdx.y;
    const int wave = tid >> 5;
    float* __restrict__ h = &hist[wave * HSTRIDE];

    for (int i = tid; i < 8 * HSTRIDE; i += 256) hist[i] = 0.0f;
    __syncthreads();

    const float4* __restrict__ pA = (const float4*)(pred + (size_t)b * 2 * PIX);
    const float4* __restrict__ pB = (const float4*)(pred + (size_t)b * 2 * PIX + PIX);
    const int4*   __restrict__ mm = (const int4*)(masks + (size_t)b * PIX);

    const int base = blockIdx.x * 256 + tid;
#pragma unroll 2
    for (int it = 0; it < ITERS; ++it) {
        const int q  = base + it * (GX * 256);
        int4      mk = mm[q];
        float4    a  = pA[q];
        float4    c  = pB[q];
        atomicAdd(&h[mk.x * 3 + 0], 1.0f);
        atomicAdd(&h[mk.x * 3 + 1], a.x);
        atomicAdd(&h[mk.x * 3 + 2], c.x);
        atomicAdd(&h[mk.y * 3 + 0], 1.0f);
        atomicAdd(&h[mk.y * 3 + 1], a.y);
        atomicAdd(&h[mk.y * 3 + 2], c.y);
        atomicAdd(&h[mk.z * 3 + 0], 1.0f);
        atomicAdd(&h[mk.z * 3 + 1], a.z);
        atomicAdd(&h[mk.z * 3 + 2], c.z);
        atomicAdd(&h[mk.w * 3 + 0], 1.0f);
        atomicAdd(&h[mk.w * 3 + 1], a.w);
        atomicAdd(&h[mk.w * 3 + 2], c.w);
    }
    __syncthreads();

    // reduce the 8 wave-private copies, one global atomic per nonzero entry
    for (int i = tid; i < HSTRIDE; i += 256) {
        float v = 0.0f;
#pragma unroll
        for (int w = 0; w < 8; ++w) v += hist[w * HSTRIDE + i];
        if (v != 0.0f) atomicAdd(&sums[b * HSTRIDE + i], v);
    }
}

// ---------------- pass 2: smooth-L1 vs segment mean, block partial sums ------
__global__ __launch_bounds__(256) void cc_loss(const float* __restrict__ pred,
                                               const int* __restrict__ masks,
                                               const float* __restrict__ sums,
                                               float* __restrict__ partials) {
    __shared__ float meanA[K_SEG];
    __shared__ float meanB[K_SEG];
    __shared__ float colsum[8 * 16];  // 8 waves x 16 column sums

    const int tid = threadIdx.x;
    const int b   = blockIdx.y;

    if (tid < K_SEG) {
        const float* s = sums + b * HSTRIDE + tid * 3;
        float inv = 1.0f / (s[0] + 1e-8f);
        meanA[tid] = s[1] * inv;
        meanB[tid] = s[2] * inv;
    }
    __syncthreads();

    const float4* __restrict__ pA = (const float4*)(pred + (size_t)b * 2 * PIX);
    const float4* __restrict__ pB = (const float4*)(pred + (size_t)b * 2 * PIX + PIX);
    const int4*   __restrict__ mm = (const int4*)(masks + (size_t)b * PIX);

    float acc0 = 0.0f, acc1 = 0.0f;
    const int base = blockIdx.x * 256 + tid;
#pragma unroll 2
    for (int it = 0; it < ITERS; ++it) {
        const int q  = base + it * (GX * 256);
        int4      mk = mm[q];
        float4    a  = pA[q];
        float4    c  = pB[q];
        acc0 += smooth_l1(a.x - meanA[mk.x]) + smooth_l1(a.y - meanA[mk.y]) +
                smooth_l1(a.z - meanA[mk.z]) + smooth_l1(a.w - meanA[mk.w]);
        acc1 += smooth_l1(c.x - meanB[mk.x]) + smooth_l1(c.y - meanB[mk.y]) +
                smooth_l1(c.z - meanB[mk.z]) + smooth_l1(c.w - meanB[mk.w]);
    }

    // Wave-level reduction with V_WMMA_F32_16X16X4_F32 (exact fp32):
    // A = ones(16x4); B vgpr0 = acc0, vgpr1 = acc1 so
    // B[0][n]=acc0_n, B[1][n]=acc1_n, B[2][n]=acc0_{n+16}, B[3][n]=acc1_{n+16}.
    // D[0][n] = acc0_n + acc1_n + acc0_{n+16} + acc1_{n+16}  (lanes 0..15, vgpr0)
    // Fixed trip count above => EXEC is all-ones here, as WMMA requires.
    v2f amat;
    amat[0] = 1.0f;
    amat[1] = 1.0f;
    v2f bmat;
    bmat[0] = acc0;
    bmat[1] = acc1;
    v8f cmat = {};
    v8f dmat = __builtin_amdgcn_wmma_f32_16x16x4_f32(
        /*neg_a=*/false, amat, /*neg_b=*/false, bmat,
        /*c_mod=*/(short)0, cmat, /*reuse_a=*/false, /*reuse_b=*/false);

    const int wave = tid >> 5;
    const int lane = tid & 31;
    if (lane < 16) colsum[wave * 16 + lane] = dmat[0];
    __syncthreads();

    if (tid == 0) {
        float t = 0.0f;
        for (int i = 0; i < 8 * 16; ++i) t += colsum[i];
        partials[blockIdx.y * GX + blockIdx.x] = t;  // one fixed slot per block
    }
}

// ---------------- pass 3: deterministic ordered final sum --------------------
__global__ void cc_finalize(const float* __restrict__ partials, int n,
                            float* __restrict__ out) {
    if (blockIdx.x == 0 && threadIdx.x == 0) {
        float t = 0.0f;
        for (int i = 0; i < n; ++i) t += partials[i];
        float loss = t / 16777216.0f;  // mean over B*C*H*W
        if (loss != loss) loss = 0.0f; // NaN -> 0
        out[0] = loss;
    }
}

extern "C" void kernel_launch(void* const* d_in, const int* in_sizes, int n_in,
                              void* d_out, int out_size, void* d_ws, size_t ws_size,
                              hipStream_t stream) {
    const float* pred  = (const float*)d_in[0];
    // d_in[1] = ab_gt: unused by the reference computation
    const int*   masks = (const int*)d_in[2];

    float* ws       = (float*)d_ws;
    float* sums     = ws;               // B*K*3 = 1536 floats
    float* partials = ws + 1536;        // GX*BATCH = 1024 floats
    float* out      = (float*)d_out;

    cc_zero_ws<<<dim3((BATCH * HSTRIDE + 255) / 256), dim3(256), 0, stream>>>(sums);
    cc_seg_sums<<<dim3(GX, BATCH), dim3(256), 0, stream>>>(pred, masks, sums);
    cc_loss<<<dim3(GX, BATCH), dim3(256), 0, stream>>>(pred, masks, sums, partials);
    cc_finalize<<<1, 64, 0, stream>>>(partials, GX * BATCH, out);
}